// GELU228_23648089932054
// MI455X (gfx1250) — compile-verified
//
#include <hip/hip_runtime.h>
#include <hip/hip_bf16.h>
#include <math.h>

typedef __attribute__((ext_vector_type(2))) float v2f;
typedef __attribute__((ext_vector_type(8))) float v8f;

#define DDIM 2048
#define PAD  4
#define DP   (DDIM + PAD)
#define NBUF 16
#define ROWS 16          // tokens per tile
#define NW   8           // waves per block (256 threads, wave32)
#define KCHUNK (DDIM / NW)   // 256 K per wave

__device__ __forceinline__ float gelu_tanh(float x) {
    const float c0 = 0.7978845608028654f;   // sqrt(2/pi)
    float x3 = x * x * x;
    float t = tanhf(c0 * (x + 0.044715f * x3));
    return 0.5f * x * (1.0f + t);
}

// Kernel 1: buf_n = L2-normalize(train_buf) rows -> workspace (16 x 2048 f32)
__global__ void norm_buf_kernel(const float* __restrict__ buf,
                                float* __restrict__ bufn) {
    __shared__ float red[256];
    const int r = blockIdx.x;
    const float* row = buf + (size_t)r * DDIM;
    float ss = 0.0f;
    for (int c = threadIdx.x; c < DDIM; c += 256) {
        float v = row[c];
        ss += v * v;
    }
    red[threadIdx.x] = ss;
    __syncthreads();
    for (int s = 128; s > 0; s >>= 1) {
        if (threadIdx.x < s) red[threadIdx.x] += red[threadIdx.x + s];
        __syncthreads();
    }
    const float inv = 1.0f / fmaxf(sqrtf(red[0]), 1e-12f);
    for (int c = threadIdx.x; c < DDIM; c += 256)
        bufn[(size_t)r * DDIM + c] = row[c] * inv;
}

// Fused: gelu -> LDS, row sumsq, WMMA f32 16x16x4 sims, mask+max, gate, y*gate
__global__ void fused_gelu_gate_kernel(const float* __restrict__ x,
                                       const float* __restrict__ bufn,
                                       const unsigned char* __restrict__ mask,
                                       const float* __restrict__ p_log_tau,
                                       const float* __restrict__ p_log_blend,
                                       float* __restrict__ out) {
    extern __shared__ float smem[];
    float* yT    = smem;                    // ROWS * DP
    float* part  = yT + ROWS * DP;          // NW * 8 * 32
    float* simss = part + NW * 8 * 32;      // 16 * 16
    float* norms = simss + 256;             // 16
    float* gateL = norms + 16;              // 16

    const int tid  = threadIdx.x;
    const int wave = tid >> 5;
    const int lane = tid & 31;
    const size_t tile = blockIdx.x;
    const float* xt = x   + tile * ROWS * (size_t)DDIM;
    float*       ot = out + tile * ROWS * (size_t)DDIM;

    // ---- Phase 1: GELU into LDS + per-row sum of squares (wave w: rows 2w,2w+1)
    {
        const int r0 = 2 * wave, r1 = 2 * wave + 1;
        const float* xr0 = xt + (size_t)r0 * DDIM;
        const float* xr1 = xt + (size_t)r1 * DDIM;
        float ss0 = 0.0f, ss1 = 0.0f;
        for (int c = lane; c < DDIM; c += 32) {
            float y0 = gelu_tanh(xr0[c]);
            float y1 = gelu_tanh(xr1[c]);
            yT[r0 * DP + c] = y0;
            yT[r1 * DP + c] = y1;
            ss0 += y0 * y0;
            ss1 += y1 * y1;
        }
        #pragma unroll
        for (int off = 16; off > 0; off >>= 1) {
            ss0 += __shfl_xor(ss0, off, 32);
            ss1 += __shfl_xor(ss1, off, 32);
        }
        if (lane == 0) {
            norms[r0] = sqrtf(ss0);
            norms[r1] = sqrtf(ss1);
        }
    }
    __syncthreads();

    // ---- Phase 2: per-wave partial C(16x16) via V_WMMA_F32_16X16X4_F32 over K-chunk
    {
        v8f acc;
        #pragma unroll
        for (int i = 0; i < 8; ++i) acc[i] = 0.0f;

        const int mn = lane & 15;              // M for A, N for B
        const int hi = (lane >> 4) << 1;       // 0 (lanes 0-15) or 2 (lanes 16-31)
        const float* yrow = yT   + mn * DP;    // unnormalized y row (divide later)
        const float* brow = bufn + (size_t)mn * DDIM; // B[k][n] = bufn[n][k]
        const int kbase = wave * KCHUNK + hi;

        #pragma unroll 4
        for (int kk = 0; kk < KCHUNK; kk += 4) {
            const int k = kbase + kk;
            v2f a, b;
            a.x = yrow[k];
            a.y = yrow[k + 1];
            b.x = brow[k];
            b.y = brow[k + 1];
            acc = __builtin_amdgcn_wmma_f32_16x16x4_f32(
                false, a, false, b, (short)0, acc, false, false);
        }
        #pragma unroll
        for (int v = 0; v < 8; ++v)
            part[(wave * 8 + v) * 32 + lane] = acc[v];
    }
    __syncthreads();

    // ---- Phase 3: reduce wave partials, normalize, mask, row-max, gate
    {
        const int m = tid >> 4, n = tid & 15;
        const int v  = m & 7;
        const int ln = n + ((m >> 3) << 4);    // C layout: lane<16 -> M=v, lane>=16 -> M=v+8
        float s = 0.0f;
        #pragma unroll
        for (int w = 0; w < NW; ++w)
            s += part[(w * 8 + v) * 32 + ln];
        s = s / fmaxf(norms[m], 1e-12f);
        if (!mask[n]) s = -1.0f;
        simss[m * 16 + n] = s;
    }
    __syncthreads();
    if (tid < ROWS) {
        float mx = -3.4028235e38f;
        #pragma unroll
        for (int n = 0; n < NBUF; ++n)
            mx = fmaxf(mx, simss[tid * 16 + n]);
        const float tau   = expf(p_log_tau[0]);
        const float lb    = p_log_blend[0];
        const float alpha = 1.0f / (1.0f + expf(-lb));
        gateL[tid] = 1.0f - alpha + alpha * expf(-tau * mx);
    }
    __syncthreads();

    // ---- Phase 4: out = y * gate (stream LDS -> HBM)
    {
        const int r0 = 2 * wave, r1 = 2 * wave + 1;
        const float g0 = gateL[r0], g1 = gateL[r1];
        for (int c = lane; c < DDIM; c += 32) {
            ot[(size_t)r0 * DDIM + c] = yT[r0 * DP + c] * g0;
            ot[(size_t)r1 * DDIM + c] = yT[r1 * DP + c] * g1;
        }
    }
}

extern "C" void kernel_launch(void* const* d_in, const int* in_sizes, int n_in,
                              void* d_out, int out_size, void* d_ws, size_t ws_size,
                              hipStream_t stream) {
    const float*         x        = (const float*)d_in[0];
    const float*         buf      = (const float*)d_in[1];
    const unsigned char* mask     = (const unsigned char*)d_in[2]; // jnp bool: 1 byte/elt
    const float*         log_tau  = (const float*)d_in[3];
    const float*         log_bl   = (const float*)d_in[4];
    float*               out      = (float*)d_out;

    float* bufn = (float*)d_ws;   // 16 * 2048 f32 = 128 KB

    const int BT    = in_sizes[0] / DDIM;   // 16384 rows
    const int tiles = BT / ROWS;            // 1024 workgroups

    norm_buf_kernel<<<NBUF, 256, 0, stream>>>(buf, bufn);

    const size_t shmem =
        (size_t)(ROWS * DP + NW * 8 * 32 + 256 + 16 + 16) * sizeof(float); // ~141 KB
    fused_gelu_gate_kernel<<<tiles, 256, shmem, stream>>>(
        x, bufn, mask, log_tau, log_bl, out);
}